// LocalFeatureAggregation_51170240364928
// MI455X (gfx1250) — compile-verified
//
#include <hip/hip_runtime.h>
#include <hip/hip_bf16.h>
#include <math.h>

// ---------------- problem constants ----------------
#define BATCH   4
#define NPTS    8192
#define SPTS    2048
#define KNN_K   16
#define INCH    64
#define OUTCH   128
#define LEAKY_S 0.1f
#define BN_EPS  1e-5f

typedef __attribute__((ext_vector_type(2))) float v2f;
typedef __attribute__((ext_vector_type(8))) float v8f;

__device__ __forceinline__ v8f wmma_f32_chunk(v8f acc, float ax, float ay, float bx, float by) {
    v2f a, b;
    a.x = ax; a.y = ay;
    b.x = bx; b.y = by;
    // D = A(16x4,f32) * B(4x16,f32) + C(16x16,f32)
    return __builtin_amdgcn_wmma_f32_16x16x4_f32(false, a, false, b, (short)0, acc, false, false);
}

__device__ __forceinline__ float leaky_f(float v) { return v >= 0.f ? v : LEAKY_S * v; }

// ---------------- K1: furthest point sampling ----------------
// one 1024-thread workgroup per batch; dist array lives in registers (8 pts/thread)
__global__ __launch_bounds__(1024) void k1_fps(const float* __restrict__ xyz,  // [B,3,N]
                                               float* __restrict__ out,        // d_out base
                                               float* __restrict__ newxyz)     // ws [B,S,3]
{
    const int b   = blockIdx.x;
    const int tid = threadIdx.x;
    const float* xb = xyz + (size_t)b * 3 * NPTS;

    float px[8], py[8], pz[8], dd[8];
#pragma unroll
    for (int j = 0; j < 8; ++j) {
        int n = tid + j * 1024;
        px[j] = xb[n]; py[j] = xb[NPTS + n]; pz[j] = xb[2 * NPTS + n];
        dd[j] = 1e10f;
    }
    __shared__ float csx, csy, csz;
    __shared__ int   ccur;
    __shared__ float wv[32];
    __shared__ int   wi[32];
    if (tid == 0) { csx = xb[0]; csy = xb[NPTS]; csz = xb[2 * NPTS]; ccur = 0; }
    __syncthreads();

    float* out_nx  = out;                                                  // [B,3,S]
    float* out_idx = out + (size_t)BATCH * 3 * SPTS + (size_t)BATCH * OUTCH * SPTS; // [B,S]

    const int wave = tid >> 5, lane = tid & 31;
    for (int i = 0; i < SPTS; ++i) {
        float qx = csx, qy = csy, qz = csz;
        int   cur = ccur;
        if (tid == 0) {
            size_t ob = (size_t)b * 3 * SPTS;
            out_nx[ob + 0 * SPTS + i] = qx;
            out_nx[ob + 1 * SPTS + i] = qy;
            out_nx[ob + 2 * SPTS + i] = qz;
            out_idx[(size_t)b * SPTS + i] = (float)cur;
            size_t nb = ((size_t)b * SPTS + i) * 3;
            newxyz[nb + 0] = qx; newxyz[nb + 1] = qy; newxyz[nb + 2] = qz;
        }
        if (i == SPTS - 1) break;

        float bv = -1.f; int bn = 0;
#pragma unroll
        for (int j = 0; j < 8; ++j) {
            float dx = px[j] - qx, dy = py[j] - qy, dz = pz[j] - qz;
            float d  = dx * dx + dy * dy + dz * dz;
            dd[j] = fminf(dd[j], d);
            int n = tid + j * 1024;
            if (dd[j] > bv || (dd[j] == bv && n < bn)) { bv = dd[j]; bn = n; }
        }
        // wave32 argmax (tie -> lower index)
#pragma unroll
        for (int off = 1; off < 32; off <<= 1) {
            float ov = __shfl_xor(bv, off, 32);
            int   oi = __shfl_xor(bn, off, 32);
            if (ov > bv || (ov == bv && oi < bn)) { bv = ov; bn = oi; }
        }
        __syncthreads();
        if (lane == 0) { wv[wave] = bv; wi[wave] = bn; }
        __syncthreads();
        if (tid < 32) {
            bv = wv[tid]; bn = wi[tid];
#pragma unroll
            for (int off = 1; off < 32; off <<= 1) {
                float ov = __shfl_xor(bv, off, 32);
                int   oi = __shfl_xor(bn, off, 32);
                if (ov > bv || (ov == bv && oi < bn)) { bv = ov; bn = oi; }
            }
            if (tid == 0) {
                ccur = bn;
                csx = xb[bn]; csy = xb[NPTS + bn]; csz = xb[2 * NPTS + bn];
            }
        }
        __syncthreads();
    }
}

// ---------------- K2: KNN (top-16) ----------------
// one 256-thread workgroup per (b,s); per-thread sorted top-16 in LDS, then 16 selection rounds
__global__ __launch_bounds__(256) void k2_knn(const float* __restrict__ xyz,    // [B,3,N]
                                              const float* __restrict__ newxyz, // [B,S,3]
                                              int* __restrict__ knn_out)        // [B,S,16]
{
    __shared__ float ld[256 * 16];
    __shared__ int   li[256 * 16];
    __shared__ float rv[256];
    __shared__ int   ri[256];
    __shared__ int   rt[256];

    const int bs = blockIdx.x;
    const int b  = bs / SPTS;
    const int tid = threadIdx.x;
    const float* xb = xyz + (size_t)b * 3 * NPTS;
    const float qx = newxyz[(size_t)bs * 3 + 0];
    const float qy = newxyz[(size_t)bs * 3 + 1];
    const float qz = newxyz[(size_t)bs * 3 + 2];

    float* myd = &ld[tid * 16];
    int*   myi = &li[tid * 16];
#pragma unroll
    for (int j = 0; j < 16; ++j) { myd[j] = 3.4e38f; myi[j] = NPTS; }

    for (int n = tid; n < NPTS; n += 256) {
        float dx = xb[n] - qx, dy = xb[NPTS + n] - qy, dz = xb[2 * NPTS + n] - qz;
        float d = dx * dx + dy * dy + dz * dz;
        if (d < myd[15] || (d == myd[15] && n < myi[15])) {
            int p = 15;
            while (p > 0 && (myd[p - 1] > d || (myd[p - 1] == d && myi[p - 1] > n))) {
                myd[p] = myd[p - 1]; myi[p] = myi[p - 1]; --p;
            }
            myd[p] = d; myi[p] = n;
        }
    }
    __syncthreads();

    int head = 0;
    for (int t = 0; t < KNN_K; ++t) {
        float v; int vi;
        if (head < 16) { v = myd[head]; vi = myi[head]; }
        else           { v = 3.5e38f;   vi = NPTS; }
        rv[tid] = v; ri[tid] = vi; rt[tid] = tid;
        __syncthreads();
        for (int sst = 128; sst > 0; sst >>= 1) {
            if (tid < sst) {
                float ov = rv[tid + sst]; int oi = ri[tid + sst];
                if (ov < rv[tid] || (ov == rv[tid] && oi < ri[tid])) {
                    rv[tid] = ov; ri[tid] = oi; rt[tid] = rt[tid + sst];
                }
            }
            __syncthreads();
        }
        int widx = ri[0], wthr = rt[0];
        if (tid == 0) knn_out[(size_t)bs * KNN_K + t] = widx;
        if (tid == wthr) head++;
        __syncthreads();
    }
}

// ---------------- Wcat build: fused [W1(9x64)->d0..63 | W0(64x64)->d64..127] as 80x128 ----------------
__global__ void k_wcat(const float* __restrict__ W0, const float* __restrict__ W1,
                       float* __restrict__ wcat)
{
    int i = blockIdx.x * blockDim.x + threadIdx.x;
    if (i >= 80 * 128) return;
    int kk = i >> 7, d = i & 127;
    float v = 0.f;
    if (kk < 9 && d < 64)           v = W1[kk * 64 + d];
    else if (kk >= 16 && d >= 64)   v = W0[(kk - 16) * 64 + (d - 64)];
    wcat[i] = v;
}

// ---------------- K3: fused MLP0 + LSE GEMM (16x80 x 80x128) with f32 WMMA ----------------
// one workgroup per (b,s): the 16 neighbors ARE the WMMA M-tile. 8 waves cover N=128.
__global__ __launch_bounds__(256) void k3_gemm01(const float* __restrict__ xyz,
                                                 const float* __restrict__ points,  // [B,64,N]
                                                 const float* __restrict__ newxyz,  // [B,S,3]
                                                 const int*   __restrict__ knn,     // [B,S,16]
                                                 const float* __restrict__ wcat,    // [80,128]
                                                 float* __restrict__ pre,           // [B*S*16,128]
                                                 float* __restrict__ sum0,
                                                 float* __restrict__ sumsq0)
{
    __shared__ float A[16][84];   // stride 84 to dodge bank conflicts
    const int bs  = blockIdx.x;
    const int b   = bs / SPTS;
    const int tid = threadIdx.x;

    const float qx = newxyz[(size_t)bs * 3 + 0];
    const float qy = newxyz[(size_t)bs * 3 + 1];
    const float qz = newxyz[(size_t)bs * 3 + 2];

    for (int i = tid; i < 16 * 84; i += 256) A[i / 84][i % 84] = 0.f;
    __syncthreads();

    const int k  = tid >> 4;      // 0..15 neighbor
    const int t2 = tid & 15;
    const int n  = knn[(size_t)bs * KNN_K + k];
    const float* xb = xyz + (size_t)b * 3 * NPTS;
    const float gx = xb[n], gy = xb[NPTS + n], gz = xb[2 * NPTS + n];
    if (t2 == 0) {
        A[k][0] = qx;      A[k][1] = qy;      A[k][2] = qz;
        A[k][3] = gx;      A[k][4] = gy;      A[k][5] = gz;
        A[k][6] = gx - qx; A[k][7] = gy - qy; A[k][8] = gz - qz;
    }
    const float* pb = points + (size_t)b * INCH * NPTS;
    for (int c = t2; c < INCH; c += 16) A[k][16 + c] = pb[(size_t)c * NPTS + n];
    __syncthreads();

    const int wave = tid >> 5, lane = tid & 31, half = lane >> 4, lm = lane & 15;
    const int n0 = wave * 16;
    v8f acc = {};
#pragma unroll
    for (int kk = 0; kk < 20; ++kk) {   // Kdim = 80
        int kb = kk * 4 + half * 2;
        acc = wmma_f32_chunk(acc, A[lm][kb], A[lm][kb + 1],
                             wcat[(kb + 0) * 128 + n0 + lm],
                             wcat[(kb + 1) * 128 + n0 + lm]);
    }
    const size_t rowbase = (size_t)bs * 16;
    float s1 = 0.f, s2 = 0.f;
#pragma unroll
    for (int r = 0; r < 8; ++r) {
        int m = half * 8 + r;
        float v = acc[r];
        pre[(rowbase + m) * 128 + n0 + lm] = v;
        s1 += v; s2 += v * v;
    }
    s1 += __shfl_xor(s1, 16, 32);
    s2 += __shfl_xor(s2, 16, 32);
    if (half == 0) {
        atomicAdd(&sum0[n0 + lm], s1);
        atomicAdd(&sumsq0[n0 + lm], s2);
    }
}

// ---------------- BN finalize: fold mean/var + gamma/beta into scale/shift ----------------
__global__ void k_finalize(const float* __restrict__ sum, const float* __restrict__ sumsq,
                           const float* __restrict__ gA, const float* __restrict__ beA,
                           const float* __restrict__ gB, const float* __restrict__ beB,
                           int split, float inv_n,
                           float* __restrict__ scale, float* __restrict__ shift)
{
    int c = threadIdx.x;  // 128 channels
    float mean = sum[c] * inv_n;
    float var  = sumsq[c] * inv_n - mean * mean;
    float g, be;
    if (c < split) { g = gA[c]; be = beA[c]; }
    else           { g = gB[c - split]; be = beB[c - split]; }
    float sc = g * rsqrtf(var + BN_EPS);
    scale[c] = sc;
    shift[c] = be - mean * sc;
}

// ---------------- K5: attention (L^T x Ws via WMMA, leaky+softmax over k, weighted sum) ----------------
__global__ __launch_bounds__(256) void k5_attn(const float* __restrict__ pre,    // [B*S*16,128]
                                               const float* __restrict__ scale0,
                                               const float* __restrict__ shift0,
                                               const float* __restrict__ Wsm,    // [128,128]
                                               float* __restrict__ feat)         // [B*S,128]
{
    __shared__ float L[16][132];  // L[k][c] = normalized+leaky lse1
    const int bs  = blockIdx.x;
    const int tid = threadIdx.x;
    const size_t rowbase = (size_t)bs * 16;
    for (int i = tid; i < 16 * 128; i += 256) {
        int kx = i >> 7, c = i & 127;
        float v = pre[(rowbase + kx) * 128 + c] * scale0[c] + shift0[c];
        L[kx][c] = leaky_f(v);
    }
    __syncthreads();

    const int wave = tid >> 5, lane = tid & 31, half = lane >> 4, lm = lane & 15;
    const int n0 = wave * 16;
    v8f acc = {};
#pragma unroll
    for (int kk = 0; kk < 32; ++kk) {   // Kdim = 128 channels
        int kb = kk * 4 + half * 2;
        acc = wmma_f32_chunk(acc, L[lm][kb], L[lm][kb + 1],
                             Wsm[(kb + 0) * 128 + n0 + lm],
                             Wsm[(kb + 1) * 128 + n0 + lm]);
    }
    // leaky -> softmax over M(=k) axis -> weighted sum with L[d,k]
    float x[8];
    float mx = -3.4e38f;
#pragma unroll
    for (int r = 0; r < 8; ++r) { float v = leaky_f(acc[r]); x[r] = v; mx = fmaxf(mx, v); }
    mx = fmaxf(mx, __shfl_xor(mx, 16, 32));
    float se = 0.f;
#pragma unroll
    for (int r = 0; r < 8; ++r) { x[r] = __expf(x[r] - mx); se += x[r]; }
    se += __shfl_xor(se, 16, 32);
    const float inv = 1.f / se;
    const int d = n0 + lm;
    float f = 0.f;
#pragma unroll
    for (int r = 0; r < 8; ++r) { int kx = half * 8 + r; f += x[r] * inv * L[kx][d]; }
    f += __shfl_xor(f, 16, 32);
    if (half == 0) feat[(size_t)bs * 128 + d] = f;
}

// ---------------- K6: final 128x128 GEMM over (b,s) rows with f32 WMMA ----------------
__global__ __launch_bounds__(256) void k6_gemm2(const float* __restrict__ feat,  // [B*S,128]
                                                const float* __restrict__ W2,    // [128,128]
                                                float* __restrict__ pre2,        // [B*S,128]
                                                float* __restrict__ sum2,
                                                float* __restrict__ sumsq2)
{
    __shared__ float A[16][132];
    const int tile = blockIdx.x;
    const int tid  = threadIdx.x;
    const size_t rb = (size_t)tile * 16;
    for (int i = tid; i < 16 * 128; i += 256) {
        int m = i >> 7, c = i & 127;
        A[m][c] = feat[(rb + m) * 128 + c];
    }
    __syncthreads();

    const int wave = tid >> 5, lane = tid & 31, half = lane >> 4, lm = lane & 15;
    const int n0 = wave * 16;
    v8f acc = {};
#pragma unroll
    for (int kk = 0; kk < 32; ++kk) {
        int kb = kk * 4 + half * 2;
        acc = wmma_f32_chunk(acc, A[lm][kb], A[lm][kb + 1],
                             W2[(kb + 0) * 128 + n0 + lm],
                             W2[(kb + 1) * 128 + n0 + lm]);
    }
    float s1 = 0.f, s2 = 0.f;
#pragma unroll
    for (int r = 0; r < 8; ++r) {
        int m = half * 8 + r;
        float v = acc[r];
        pre2[(rb + m) * 128 + n0 + lm] = v;
        s1 += v; s2 += v * v;
    }
    s1 += __shfl_xor(s1, 16, 32);
    s2 += __shfl_xor(s2, 16, 32);
    if (half == 0) {
        atomicAdd(&sum2[n0 + lm], s1);
        atomicAdd(&sumsq2[n0 + lm], s2);
    }
}

// ---------------- K8: final BN + leaky, transpose to [B,OUT,S] ----------------
__global__ void k8_out(const float* __restrict__ pre2, const float* __restrict__ scale2,
                       const float* __restrict__ shift2, float* __restrict__ out_np)
{
    int i = blockIdx.x * blockDim.x + threadIdx.x;  // B*S*128
    int bs = i >> 7, d = i & 127;
    int b = bs / SPTS, s = bs % SPTS;
    float v = pre2[i] * scale2[d] + shift2[d];
    out_np[((size_t)b * OUTCH + d) * SPTS + s] = leaky_f(v);
}

// ---------------- workspace layout (float units) ----------------
static const size_t OFF_NEWXYZ = 0;                                          // [B,S,3]   24576 f
static const size_t OFF_KNN    = OFF_NEWXYZ + (size_t)BATCH * SPTS * 3;      // [B,S,16]  ints
static const size_t OFF_WCAT   = OFF_KNN + (size_t)BATCH * SPTS * KNN_K;     // 80*128
static const size_t OFF_PRE    = OFF_WCAT + 80 * 128;                        // [B*S*16,128]
static const size_t OFF_FEAT   = OFF_PRE + (size_t)BATCH * SPTS * KNN_K * 128;
static const size_t OFF_PRE2   = OFF_FEAT + (size_t)BATCH * SPTS * 128;
static const size_t OFF_STAT   = OFF_PRE2 + (size_t)BATCH * SPTS * 128;      // 1024 f of stats

extern "C" void kernel_launch(void* const* d_in, const int* in_sizes, int n_in,
                              void* d_out, int out_size, void* d_ws, size_t ws_size,
                              hipStream_t stream) {
    (void)in_sizes; (void)n_in; (void)out_size; (void)ws_size;
    const float* xyz    = (const float*)d_in[0];
    const float* points = (const float*)d_in[1];
    const float* W0     = (const float*)d_in[2];
    // d_in[3] = b0 : cancelled by train-mode BN
    const float* g0     = (const float*)d_in[4];
    const float* be0    = (const float*)d_in[5];
    const float* W1     = (const float*)d_in[6];
    // d_in[7] = b1 : cancelled
    const float* g1     = (const float*)d_in[8];
    const float* be1    = (const float*)d_in[9];
    const float* Wsm    = (const float*)d_in[10];
    const float* W2     = (const float*)d_in[11];
    // d_in[12] = b2 : cancelled
    const float* g2     = (const float*)d_in[13];
    const float* be2    = (const float*)d_in[14];

    float* out = (float*)d_out;
    float* ws  = (float*)d_ws;

    float* nx_ws   = ws + OFF_NEWXYZ;
    int*   knn_ws  = (int*)(ws + OFF_KNN);
    float* wcat_ws = ws + OFF_WCAT;
    float* pre_ws  = ws + OFF_PRE;
    float* feat_ws = ws + OFF_FEAT;
    float* pre2_ws = ws + OFF_PRE2;
    float* st      = ws + OFF_STAT;
    // st layout: sum0[128] sumsq0[128] scale0[128] shift0[128] sum2[128] sumsq2[128] scale2[128] shift2[128]

    hipMemsetAsync((void*)st, 0, 1024 * sizeof(float), stream);

    k_wcat<<<40, 256, 0, stream>>>(W0, W1, wcat_ws);
    k1_fps<<<BATCH, 1024, 0, stream>>>(xyz, out, nx_ws);
    k2_knn<<<BATCH * SPTS, 256, 0, stream>>>(xyz, nx_ws, knn_ws);
    k3_gemm01<<<BATCH * SPTS, 256, 0, stream>>>(xyz, points, nx_ws, knn_ws, wcat_ws,
                                                pre_ws, st + 0, st + 128);
    k_finalize<<<1, 128, 0, stream>>>(st + 0, st + 128, g1, be1, g0, be0, 64,
                                      1.0f / (float)(BATCH * SPTS * KNN_K),
                                      st + 256, st + 384);
    k5_attn<<<BATCH * SPTS, 256, 0, stream>>>(pre_ws, st + 256, st + 384, Wsm, feat_ws);
    k6_gemm2<<<(BATCH * SPTS) / 16, 256, 0, stream>>>(feat_ws, W2, pre2_ws,
                                                      st + 512, st + 640);
    k_finalize<<<1, 128, 0, stream>>>(st + 512, st + 640, g2, be2, g2, be2, 128,
                                      1.0f / (float)(BATCH * SPTS),
                                      st + 768, st + 896);
    k8_out<<<(BATCH * SPTS * OUTCH) / 256, 256, 0, stream>>>(pre2_ws, st + 768, st + 896,
                                                             out + (size_t)BATCH * 3 * SPTS);
}